// NeuralMemory_7756710936802
// MI455X (gfx1250) — compile-verified
//
#include <hip/hip_runtime.h>
#include <hip/hip_bf16.h>
#include <math.h>

// ---------------------------------------------------------------------------
// Titans neural-memory "store" step for MI455X (gfx1250, wave32, WMMA f32).
// Bandwidth-bound: gradients written straight to d_out, scans run in-place.
// LDS tiles staged with GLOBAL_LOAD_ASYNC_TO_LDS_B128 (ASYNCcnt tracked).
// ---------------------------------------------------------------------------

#define DIMX   512
#define HEADS  4
#define DH     128
#define CHUNKC 16
#define NCH    128            // S / CHUNK
#define SEQL   2048
#define BATCH  2
#define BHN    (BATCH*HEADS)  // 8
#define MAXLR  0.01f
#define EPSR   1e-6f

typedef float v2f __attribute__((ext_vector_type(2)));
typedef float v8f __attribute__((ext_vector_type(8)));

// ---- gfx1250 async-copy-to-LDS support (guarded; falls back to plain copy) --
#if defined(__AMDGCN__) &&                                          \
    __has_builtin(__builtin_amdgcn_global_load_async_to_lds_b128) && \
    __has_builtin(__builtin_amdgcn_s_wait_asynccnt)
#define USE_ASYNC_LDS 1
#else
#define USE_ASYNC_LDS 0
#endif

#if USE_ASYNC_LDS
// Signature discovered via compiler diagnostics:
//   arg0: int __attribute__((vector_size(16))) *              (global src)
//   arg1: int __attribute__((vector_size(16))) __shared__ *   (LDS dst)
//   arg2/arg3: immediate int offset / cpol
typedef int b128_t __attribute__((vector_size(16)));
typedef __attribute__((address_space(3))) b128_t lds_b128_t;

// Flat pointers to LDS carry the LDS byte offset in their low 32 bits
// (aperture base has zero low bits), so integer truncation recovers the
// addrspace(3) pointer.
__device__ __forceinline__ lds_b128_t* to_lds(const void* p) {
  return (lds_b128_t*)(uintptr_t)(uint32_t)(uintptr_t)p;
}
__device__ __forceinline__ b128_t* glb_b128(const void* p) {
  return (b128_t*)(uintptr_t)p;   // also drops const
}
#endif

__device__ __forceinline__ float sigmoidf_(float x) {
  return 1.0f / (1.0f + __expf(-x));
}

__device__ __forceinline__ v8f wmma_f32(v2f a, v2f b, v8f c) {
  // D = A(16x4,f32) * B(4x16,f32) + C(16x16,f32)
  return __builtin_amdgcn_wmma_f32_16x16x4_f32(false, a, false, b,
                                               (short)0, c, false, false);
}

// ---------------------------------------------------------------------------
// Kernel 1: RMSNorm (with scale) + per-token adaptive lr = sigmoid(x@W_step)*max_lr
// grid = B*S blocks, 256 threads (each thread owns dims tid and tid+256)
// ---------------------------------------------------------------------------
__global__ void __launch_bounds__(256) k_rmsnorm_lr(
    const float* __restrict__ seq, const float* __restrict__ scale,
    const float* __restrict__ Wstep, float* __restrict__ xn,
    float* __restrict__ lr)
{
  const int t   = blockIdx.x;        // token index 0..B*S-1
  const int tid = threadIdx.x;
  const int d0 = tid, d1 = tid + 256;

  __shared__ float red[256];
  __shared__ float red4[4][256];

  const float x0 = seq[(size_t)t * DIMX + d0];
  const float x1 = seq[(size_t)t * DIMX + d1];
  red[tid] = x0 * x0 + x1 * x1;
  __syncthreads();
  for (int off = 128; off > 0; off >>= 1) {
    if (tid < off) red[tid] += red[tid + off];
    __syncthreads();
  }
  const float inv = rsqrtf(red[0] * (1.0f / DIMX) + EPSR);
  const float n0 = x0 * inv * scale[d0];
  const float n1 = x1 * inv * scale[d1];
  xn[(size_t)t * DIMX + d0] = n0;
  xn[(size_t)t * DIMX + d1] = n1;

#pragma unroll
  for (int h = 0; h < 4; ++h)
    red4[h][tid] = n0 * Wstep[d0 * 4 + h] + n1 * Wstep[d1 * 4 + h];
  __syncthreads();
  for (int off = 128; off > 0; off >>= 1) {
    if (tid < off) {
#pragma unroll
      for (int h = 0; h < 4; ++h) red4[h][tid] += red4[h][tid + off];
    }
    __syncthreads();
  }
  if (tid < 4) {
    const int b = t / SEQL, s = t % SEQL;
    lr[((size_t)(b * HEADS + tid)) * SEQL + s] = MAXLR * sigmoidf_(red4[tid][0]);
  }
}

// ---------------------------------------------------------------------------
// Kernel 2: per-chunk mean -> mom/dec sigmoids.  grid = B*NCH blocks.
// ---------------------------------------------------------------------------
__global__ void __launch_bounds__(256) k_chunk_stats(
    const float* __restrict__ xn, const float* __restrict__ Wmom,
    const float* __restrict__ Wdec, float* __restrict__ mom,
    float* __restrict__ dec)
{
  const int bn = blockIdx.x;
  const int b = bn / NCH, n = bn % NCH;
  const int tid = threadIdx.x;
  const int d0 = tid, d1 = tid + 256;

  const float* base = xn + ((size_t)b * SEQL + (size_t)n * CHUNKC) * DIMX;
  float m0 = 0.f, m1 = 0.f;
  for (int c = 0; c < CHUNKC; ++c) {
    m0 += base[c * DIMX + d0];
    m1 += base[c * DIMX + d1];
  }
  m0 *= (1.0f / CHUNKC);
  m1 *= (1.0f / CHUNKC);

  __shared__ float red8[8][256];
#pragma unroll
  for (int h = 0; h < 4; ++h) {
    red8[h][tid]     = m0 * Wmom[d0 * 4 + h] + m1 * Wmom[d1 * 4 + h];
    red8[4 + h][tid] = m0 * Wdec[d0 * 4 + h] + m1 * Wdec[d1 * 4 + h];
  }
  __syncthreads();
  for (int off = 128; off > 0; off >>= 1) {
    if (tid < off) {
#pragma unroll
      for (int q = 0; q < 8; ++q) red8[q][tid] += red8[q][tid + off];
    }
    __syncthreads();
  }
  if (tid < 4) {
    mom[((size_t)(b * HEADS + tid)) * NCH + n] = sigmoidf_(red8[tid][0]);
    dec[((size_t)(b * HEADS + tid)) * NCH + n] = sigmoidf_(red8[4 + tid][0]);
  }
}

// ---------------------------------------------------------------------------
// Kernel 3: kv = xn @ W_kv  via V_WMMA_F32_16X16X4_F32.
// grid = (256 M-tiles, 8 col-groups), 256 threads = 8 waves, 1 tile per wave.
// A 16-token M-tile is exactly one chunk, so the k/v scatter indices are
// block/wave-uniform: the epilogue is 8 plain strided stores.
// The 16x512 token tile is shared by all 8 waves -> staged once in LDS via
// async-to-LDS copies; B fragments stream from L2-resident W_kv.
// ---------------------------------------------------------------------------
__global__ void __launch_bounds__(256) k_kv_proj(
    const float* __restrict__ xn, const float* __restrict__ Wkv,
    float* __restrict__ kbuf, float* __restrict__ vbuf)
{
  __shared__ float s_x[16 * DIMX];               // 32 KB token tile

  const int tid  = threadIdx.x;
  const int lane = tid & 31;
  const int wave = tid >> 5;
  const int half = lane >> 4;
  const int lm   = lane & 15;
  const int row0 = blockIdx.x * 16;              // token tile base == chunk base
  const int ct   = blockIdx.y * 8 + wave;        // column tile 0..63
  const int col0 = ct * 16;

  const float* src = xn + (size_t)row0 * DIMX;
#if USE_ASYNC_LDS
  for (int i = tid; i < (16 * DIMX) / 4; i += 256)
    __builtin_amdgcn_global_load_async_to_lds_b128(glb_b128(src + i * 4),
                                                   to_lds(s_x + i * 4), 0, 0);
  __builtin_amdgcn_s_wait_asynccnt(0);
#else
  for (int i = tid; i < 16 * DIMX; i += 256) s_x[i] = src[i];
#endif
  __syncthreads();

  const float* bcol = Wkv + col0 + lm;
  v8f acc = {};
  for (int k0 = 0; k0 < DIMX; k0 += 4) {
    v2f a, b;
    a.x = s_x[lm * DIMX + k0 + 2 * half + 0];
    a.y = s_x[lm * DIMX + k0 + 2 * half + 1];
    b.x = bcol[(size_t)(k0 + 2 * half + 0) * 1024];
    b.y = bcol[(size_t)(k0 + 2 * half + 1) * 1024];
    acc = wmma_f32(a, b, acc);
  }

  // Block-uniform chunk coordinates (tile == chunk).
  const int b_ = row0 / SEQL;
  const int n  = (row0 % SEQL) / CHUNKC;
  // Wave-uniform k/v destination (512-column boundary is tile-aligned).
  const bool isv = (col0 >= 512);
  const int  jc  = isv ? (col0 - 512) : col0;
  const int  h   = jc >> 7;
  float* dst = (isv ? vbuf : kbuf) +
               (((size_t)(b_ * HEADS + h) * NCH + n) * CHUNKC) * DH +
               (size_t)(8 * half) * DH + (jc & 127) + lm;
#pragma unroll
  for (int r = 0; r < 8; ++r)
    dst[(size_t)r * DH] = acc[r];   // token c = r + 8*half, dim d = (jc&127)+lm
}

// ---------------------------------------------------------------------------
// Kernel 4: per-chunk MLP gradients (all matmuls via WMMA f32), writes -g to out.
// grid = BH*NCH blocks (1024), 256 threads = 8 waves.
// k/v chunk tiles staged via async-to-LDS; w0/w1 stream from L2.
// ---------------------------------------------------------------------------
__global__ void __launch_bounds__(256) k_chunk_grad(
    const float* __restrict__ kbuf, const float* __restrict__ vbuf,
    const float* __restrict__ lrbuf, const float* __restrict__ w0,
    const float* __restrict__ w1, float* __restrict__ out)
{
  const int blk = blockIdx.x;
  const int bh = blk / NCH, n = blk % NCH;
  const int tid  = threadIdx.x;
  const int lane = tid & 31, wave = tid >> 5;
  const int half = lane >> 4, lm = lane & 15;

  __shared__ float s_k[CHUNKC * DH];
  __shared__ float s_v[CHUNKC * DH];
  __shared__ float s_h[CHUNKC * DH];
  __shared__ float s_a[CHUNKC * DH];
  __shared__ float s_dp[CHUNKC * DH];
  __shared__ float s_dh[CHUNKC * DH];
  __shared__ float s_lr[CHUNKC];

  const float* kg = kbuf + (size_t)blk * (CHUNKC * DH);
  const float* vg = vbuf + (size_t)blk * (CHUNKC * DH);
#if USE_ASYNC_LDS
  for (int i = tid; i < (CHUNKC * DH) / 4; i += 256) {
    __builtin_amdgcn_global_load_async_to_lds_b128(glb_b128(kg + i * 4),
                                                   to_lds(s_k + i * 4), 0, 0);
    __builtin_amdgcn_global_load_async_to_lds_b128(glb_b128(vg + i * 4),
                                                   to_lds(s_v + i * 4), 0, 0);
  }
#else
  for (int i = tid; i < CHUNKC * DH; i += 256) {
    s_k[i] = kg[i];
    s_v[i] = vg[i];
  }
#endif
  if (tid < CHUNKC) s_lr[tid] = lrbuf[(size_t)bh * SEQL + n * CHUNKC + tid];

  const float* w0g = w0 + (size_t)bh * DH * DH;
  const float* w1g = w1 + (size_t)bh * DH * DH;
  __builtin_prefetch(w0g, 0, 0);   // global_prefetch_b8: warm caches for weights
  __builtin_prefetch(w1g, 0, 0);
#if USE_ASYNC_LDS
  __builtin_amdgcn_s_wait_asynccnt(0);
#endif
  __syncthreads();

  const int col0 = wave * 16;

  // ---- h = k @ w0 ; a = silu(h) ----
  {
    v8f acc = {};
    for (int k0 = 0; k0 < DH; k0 += 4) {
      v2f a, b;
      a.x = s_k[lm * DH + k0 + 2 * half + 0];
      a.y = s_k[lm * DH + k0 + 2 * half + 1];
      b.x = w0g[(k0 + 2 * half + 0) * DH + col0 + lm];
      b.y = w0g[(k0 + 2 * half + 1) * DH + col0 + lm];
      acc = wmma_f32(a, b, acc);
    }
#pragma unroll
    for (int r = 0; r < 8; ++r) {
      const int m = r + 8 * half, c = col0 + lm;
      const float hv = acc[r];
      const float sg = sigmoidf_(hv);
      s_h[m * DH + c] = hv;
      s_a[m * DH + c] = hv * sg;
    }
  }
  __syncthreads();

  // ---- pred = a @ w1 ; dpred = (2/DH) * lr * (pred - v) ----
  {
    v8f acc = {};
    for (int k0 = 0; k0 < DH; k0 += 4) {
      v2f a, b;
      a.x = s_a[lm * DH + k0 + 2 * half + 0];
      a.y = s_a[lm * DH + k0 + 2 * half + 1];
      b.x = w1g[(k0 + 2 * half + 0) * DH + col0 + lm];
      b.y = w1g[(k0 + 2 * half + 1) * DH + col0 + lm];
      acc = wmma_f32(a, b, acc);
    }
#pragma unroll
    for (int r = 0; r < 8; ++r) {
      const int m = r + 8 * half, c = col0 + lm;
      s_dp[m * DH + c] = (2.0f / DH) * s_lr[m] * (acc[r] - s_v[m * DH + c]);
    }
  }
  __syncthreads();

  // ---- da = dpred @ w1^T ; dh_pre = da * silu'(h) ----
  {
    v8f acc = {};
    for (int k0 = 0; k0 < DH; k0 += 4) {
      v2f a, b;
      a.x = s_dp[lm * DH + k0 + 2 * half + 0];
      a.y = s_dp[lm * DH + k0 + 2 * half + 1];
      b.x = w1g[(col0 + lm) * DH + k0 + 2 * half + 0];   // w1^T
      b.y = w1g[(col0 + lm) * DH + k0 + 2 * half + 1];
      acc = wmma_f32(a, b, acc);
    }
#pragma unroll
    for (int r = 0; r < 8; ++r) {
      const int m = r + 8 * half, c = col0 + lm;
      const float hv = s_h[m * DH + c];
      const float sg = sigmoidf_(hv);
      s_dh[m * DH + c] = acc[r] * (sg * (1.0f + hv * (1.0f - sg)));
    }
  }
  __syncthreads();

  // ---- g0 = k^T @ dh_pre ; g1 = a^T @ dpred ; out = -g (surprises) ----
  // 128 16x16 tiles (2 mats * 8 * 8) over 8 waves => 16 tiles per wave, K=16.
  float* out0 = out + ((size_t)bh * NCH + n) * (size_t)(DH * DH);
  for (int t = 0; t < 16; ++t) {
    const int id  = wave * 16 + t;   // 0..127
    const int mat = id >> 6;         // 0 -> g0, 1 -> g1
    const int ti  = id & 63;
    const int i = ti >> 3, j = ti & 7;
    const float* A  = mat ? s_a  : s_k;   // [16,128], consumed transposed
    const float* Bm = mat ? s_dp : s_dh;  // [16,128]
    v8f acc = {};
#pragma unroll
    for (int k0 = 0; k0 < CHUNKC; k0 += 4) {
      v2f a, b;
      a.x = A[(k0 + 2 * half + 0) * DH + i * 16 + lm];
      a.y = A[(k0 + 2 * half + 1) * DH + i * 16 + lm];
      b.x = Bm[(k0 + 2 * half + 0) * DH + j * 16 + lm];
      b.y = Bm[(k0 + 2 * half + 1) * DH + j * 16 + lm];
      acc = wmma_f32(a, b, acc);
    }
    float* dst = out0 + (size_t)mat * ((size_t)BHN * NCH * DH * DH) +
                 (size_t)(i * 16 + 8 * half) * DH + j * 16 + lm;
#pragma unroll
    for (int r = 0; r < 8; ++r)
      dst[(size_t)r * DH] = -acc[r];
  }
}

// ---------------------------------------------------------------------------
// Kernel 5: fused momentum + decay linear scans, in place on d_out.
// One thread per (mat, bh, element): 262144 threads, serial over 128 chunks.
// Consecutive threads touch consecutive addresses each step (coalesced).
// ---------------------------------------------------------------------------
__global__ void __launch_bounds__(256) k_scan(
    float* __restrict__ out, const float* __restrict__ mom,
    const float* __restrict__ dec)
{
  const int gid = blockIdx.x * blockDim.x + threadIdx.x;
  const int e   = gid & (DH * DH - 1);
  const int rem = gid / (DH * DH);
  const int bh  = rem & (BHN - 1);
  const int mat = rem >> 3;

  float* base = out + ((size_t)(mat * BHN + bh) * NCH) * (size_t)(DH * DH) + e;
  const float* mo = mom + (size_t)bh * NCH;
  const float* de = dec + (size_t)bh * NCH;

  float m = 0.f, u = 0.f;
  for (int n = 0; n < NCH; ++n) {
    const float s = base[(size_t)n * DH * DH];
    m = mo[n] * m + s;                 // momentum scan
    u = (1.0f - de[n]) * u + m;        // decay scan
    base[(size_t)n * DH * DH] = u;
  }
}

// ---------------------------------------------------------------------------
extern "C" void kernel_launch(void* const* d_in, const int* in_sizes, int n_in,
                              void* d_out, int out_size, void* d_ws,
                              size_t ws_size, hipStream_t stream)
{
  const float* seq   = (const float*)d_in[0];
  const float* scale = (const float*)d_in[1];
  const float* Wkv   = (const float*)d_in[2];
  const float* Wstep = (const float*)d_in[3];
  const float* Wmom  = (const float*)d_in[4];
  const float* Wdec  = (const float*)d_in[5];
  const float* w0    = (const float*)d_in[6];
  const float* w1    = (const float*)d_in[7];
  float* out = (float*)d_out;

  // workspace layout (floats): ~25.2 MB total
  float* ws   = (float*)d_ws;
  float* xn   = ws;                                   // B*S*DIM   = 2,097,152
  float* kbuf = xn   + (size_t)BATCH * SEQL * DIMX;   // 2,097,152
  float* vbuf = kbuf + (size_t)2097152;               // 2,097,152
  float* lr   = vbuf + (size_t)2097152;               // BH*S      = 16,384
  float* mom  = lr   + (size_t)BHN * SEQL;            // BH*NCH    = 1,024
  float* dec  = mom  + (size_t)BHN * NCH;             // 1,024

  k_rmsnorm_lr<<<BATCH * SEQL, 256, 0, stream>>>(seq, scale, Wstep, xn, lr);
  k_chunk_stats<<<BATCH * NCH, 256, 0, stream>>>(xn, Wmom, Wdec, mom, dec);
  k_kv_proj<<<dim3(BATCH * SEQL / 16, 8), 256, 0, stream>>>(xn, Wkv, kbuf, vbuf);
  k_chunk_grad<<<BHN * NCH, 256, 0, stream>>>(kbuf, vbuf, lr, w0, w1, out);
  k_scan<<<(2 * BHN * DH * DH) / 256, 256, 0, stream>>>(out, mom, dec);
}